// GCN_90718299226513
// MI455X (gfx1250) — compile-verified
//
#include <hip/hip_runtime.h>
#include <hip/hip_bf16.h>
#include <math.h>

// ---------------------------------------------------------------------------
// 3-layer GCN for MI455X (gfx1250, wave32).
//  - GEMMs use V_WMMA_F32_16X16X4_F32 (full f32 precision; GEMM is ~0.8 GFLOP
//    total, so low-precision WMMA buys nothing -- the aggregation dominates).
//  - Aggregation is L2-resident (node tables are 5.12MB << 192MB L2):
//    one wave32 per edge, 128-bit vector gather of H[src], f32 global atomics
//    into AGG[dst].
// ---------------------------------------------------------------------------

typedef __attribute__((ext_vector_type(2))) float v2f;
typedef __attribute__((ext_vector_type(8))) float v8f;

#define N_NODES 10000
#define N_EDGES 640000

// ---------------- degree / normalization ----------------------------------
__global__ void k_init_deg(float* __restrict__ deg) {
    int i = blockIdx.x * blockDim.x + threadIdx.x;
    if (i < N_NODES) deg[i] = 1.0f;            // self-loop contributes 1
}

__global__ void k_degree(const int* __restrict__ dst, float* __restrict__ deg) {
    int e = blockIdx.x * blockDim.x + threadIdx.x;
    if (e < N_EDGES) atomicAdd(&deg[dst[e]], 1.0f);
}

__global__ void k_dinv(float* __restrict__ dinv) {
    int i = blockIdx.x * blockDim.x + threadIdx.x;
    if (i < N_NODES) dinv[i] = rsqrtf(dinv[i]); // deg >= 1 always (self-loop)
}

// ---------------- WMMA GEMM: H[N x DOUT] = X[N x K] @ W[K x DOUT] ----------
// One wave32 per 16x16 output tile. K-loop: K/4 V_WMMA_F32_16X16X4_F32 ops.
// A-frag (16x4 f32): lane L -> row = tm*16 + (L&15), k = k0 + 2*(L>>4) + {0,1}
// B-frag (4x16 f32): lane L -> col = tn*16 + (L&15), same k split.
// C/D (16x16 f32, 8 VGPRs): vgpr i -> M = i + 8*(L>>4), N = L&15.
template <int K, int DOUT>
__global__ void k_gemm_wmma(const float* __restrict__ X,
                            const float* __restrict__ W,
                            float* __restrict__ H,
                            int total_tiles) {
    const int n_col_tiles = DOUT / 16;
    int wave = blockIdx.x * (blockDim.x >> 5) + (threadIdx.x >> 5);
    if (wave >= total_tiles) return;           // wave-uniform exit (EXEC all-1)
    int lane  = threadIdx.x & 31;
    int tm    = wave / n_col_tiles;
    int tn    = wave % n_col_tiles;
    int lhalf = (lane >> 4);                   // 0: k+{0,1}, 1: k+{2,3}
    int row_a = tm * 16 + (lane & 15);
    int col_b = tn * 16 + (lane & 15);

    const float* xrow = X + (size_t)row_a * K + 2 * lhalf;

    v8f c = {};
#pragma unroll
    for (int k0 = 0; k0 < K; k0 += 4) {
        v2f a = *(const v2f*)(xrow + k0);                  // 8B-aligned pair
        int kb = k0 + 2 * lhalf;
        v2f b;
        b.x = W[(size_t)kb * DOUT + col_b];
        b.y = W[(size_t)(kb + 1) * DOUT + col_b];
        c = __builtin_amdgcn_wmma_f32_16x16x4_f32(
                /*neg_a=*/false, a, /*neg_b=*/false, b,
                /*c_mod=*/(short)0, c, /*reuse_a=*/false, /*reuse_b=*/false);
    }

    int rbase = tm * 16 + (lhalf ? 8 : 0);
    int col   = tn * 16 + (lane & 15);
#pragma unroll
    for (int i = 0; i < 8; ++i)
        H[(size_t)(rbase + i) * DOUT + col] = c[i];
}

// ---------------- self-loop init: AGG[i] = dinv[i]^2 * H[i] ----------------
__global__ void k_selfloop(const float* __restrict__ H,
                           const float* __restrict__ dinv,
                           float* __restrict__ AGG, int D) {
    int t = blockIdx.x * blockDim.x + threadIdx.x;
    if (t >= N_NODES * D) return;
    float s = dinv[t / D];
    AGG[t] = s * s * H[t];
}

// ---------------- edge aggregation: AGG[dst] += norm * H[src] --------------
// One wave32 per edge; each lane owns D/32 contiguous features.
template <int D>
__global__ void k_edge_agg(const int* __restrict__ src,
                           const int* __restrict__ dst,
                           const float* __restrict__ dinv,
                           const float* __restrict__ H,
                           float* __restrict__ AGG) {
    constexpr int CH = D / 32;                  // 4 (d=128) or 2 (d=64)
    int wave = blockIdx.x * (blockDim.x >> 5) + (threadIdx.x >> 5);
    if (wave >= N_EDGES) return;
    int lane = threadIdx.x & 31;
    int s = src[wave];
    int d = dst[wave];
    float nrm = dinv[s] * dinv[d];
    const float* hrow = H   + (size_t)s * D + lane * CH;
    float*       arow = AGG + (size_t)d * D + lane * CH;
    if constexpr (CH == 4) {
        float4 v = *(const float4*)hrow;        // 16B-aligned vector gather
        atomicAdd(arow + 0, v.x * nrm);
        atomicAdd(arow + 1, v.y * nrm);
        atomicAdd(arow + 2, v.z * nrm);
        atomicAdd(arow + 3, v.w * nrm);
    } else {
        float2 v = *(const float2*)hrow;
        atomicAdd(arow + 0, v.x * nrm);
        atomicAdd(arow + 1, v.y * nrm);
    }
}

// ---------------- epilogue: out = [sigmoid](AGG + b) -----------------------
template <int D, bool SIGMOID>
__global__ void k_bias_act(const float* __restrict__ AGG,
                           const float* __restrict__ bias,
                           float* __restrict__ out) {
    int t = blockIdx.x * blockDim.x + threadIdx.x;
    if (t >= N_NODES * D) return;
    float v = AGG[t] + bias[t & (D - 1)];
    if constexpr (SIGMOID) v = 1.0f / (1.0f + __expf(-v));
    out[t] = v;
}

// ---------------------------------------------------------------------------
extern "C" void kernel_launch(void* const* d_in, const int* in_sizes, int n_in,
                              void* d_out, int out_size, void* d_ws, size_t ws_size,
                              hipStream_t stream) {
    (void)in_sizes; (void)n_in; (void)out_size; (void)ws_size;

    const float* x   = (const float*)d_in[0];
    const int*   ei  = (const int*)d_in[1];      // [2, E]: row0=src, row1=dst
    /* d_in[2] = edge_attr: unused by the reference */
    const float* W1  = (const float*)d_in[3];
    const float* b1  = (const float*)d_in[4];
    const float* W2  = (const float*)d_in[5];
    const float* b2  = (const float*)d_in[6];
    const float* W3  = (const float*)d_in[7];
    const float* b3  = (const float*)d_in[8];
    float* out = (float*)d_out;

    const int* src = ei;
    const int* dst = ei + N_EDGES;

    // workspace layout (all offsets 256B-aligned)
    const size_t NODE128 = (size_t)N_NODES * 128 * sizeof(float); // 5,120,000
    char* ws = (char*)d_ws;
    float* dinv = (float*)ws;                       // 40,000 B (pad to 64KB)
    float* H    = (float*)(ws + 65536);
    float* AGG  = (float*)(ws + 65536 + NODE128);
    float* XB   = (float*)(ws + 65536 + 2 * NODE128);

    const int B = 256;                              // 8 waves per workgroup
    const int wavesPerBlk = B / 32;

    // ---- normalization: deg -> dinv ----
    k_init_deg<<<(N_NODES + B - 1) / B, B, 0, stream>>>(dinv);
    k_degree  <<<(N_EDGES + B - 1) / B, B, 0, stream>>>(dst, dinv);
    k_dinv    <<<(N_NODES + B - 1) / B, B, 0, stream>>>(dinv);

    const int rowTiles  = N_NODES / 16;             // 625 exact
    const int tiles128  = rowTiles * (128 / 16);    // 5000
    const int tiles64   = rowTiles * (64 / 16);     // 2500
    const int edgeBlks  = (N_EDGES * 32 + B - 1) / B;

    // ---- layer 1: x -> XB ----
    k_gemm_wmma<128, 128><<<(tiles128 + wavesPerBlk - 1) / wavesPerBlk, B, 0, stream>>>(
        x, W1, H, tiles128);
    k_selfloop<<<(N_NODES * 128 + B - 1) / B, B, 0, stream>>>(H, dinv, AGG, 128);
    k_edge_agg<128><<<edgeBlks, B, 0, stream>>>(src, dst, dinv, H, AGG);
    k_bias_act<128, true><<<(N_NODES * 128 + B - 1) / B, B, 0, stream>>>(AGG, b1, XB);

    // ---- layer 2: XB -> XB ----
    k_gemm_wmma<128, 128><<<(tiles128 + wavesPerBlk - 1) / wavesPerBlk, B, 0, stream>>>(
        XB, W2, H, tiles128);
    k_selfloop<<<(N_NODES * 128 + B - 1) / B, B, 0, stream>>>(H, dinv, AGG, 128);
    k_edge_agg<128><<<edgeBlks, B, 0, stream>>>(src, dst, dinv, H, AGG);
    k_bias_act<128, true><<<(N_NODES * 128 + B - 1) / B, B, 0, stream>>>(AGG, b2, XB);

    // ---- layer 3: XB -> out (no sigmoid) ----
    k_gemm_wmma<128, 64><<<(tiles64 + wavesPerBlk - 1) / wavesPerBlk, B, 0, stream>>>(
        XB, W3, H, tiles64);
    k_selfloop<<<(N_NODES * 64 + B - 1) / B, B, 0, stream>>>(H, dinv, AGG, 64);
    k_edge_agg<64><<<edgeBlks, B, 0, stream>>>(src, dst, dinv, H, AGG);
    k_bias_act<64, false><<<(N_NODES * 64 + B - 1) / B, B, 0, stream>>>(AGG, b3, out);
}